// Model_23158463660646
// MI455X (gfx1250) — compile-verified
//
#include <hip/hip_runtime.h>
#include <hip/hip_bf16.h>

typedef float v2f __attribute__((ext_vector_type(2)));
typedef float v8f __attribute__((ext_vector_type(8)));

#define GS_N 4096
#define GS_W 96
#define GS_H 96
#define GS_TANFOV 0.36502849483f   /* tan(0.35) */

// ---------------------------------------------------------------------------
// Kernel 1: per-Gaussian preprocessing
// ---------------------------------------------------------------------------
__global__ __launch_bounds__(256) void gs_preprocess(
    const float* __restrict__ means3D,      // [N,3]
    const float* __restrict__ scales_raw,   // [N,3]
    const float* __restrict__ rot_raw,      // [N,4]
    const float* __restrict__ opac_raw,     // [N,1]
    const float* __restrict__ shs,          // [N,16,3]
    const float* __restrict__ view,         // [4,4] row-major
    const float* __restrict__ proj,         // [4,4] row-major (full proj)
    const float* __restrict__ campos,       // [3]
    float* __restrict__ depthKey,           // [N]
    float* __restrict__ coeff8,             // [N,8] power-form coefficients
    float4* __restrict__ colOp)             // [N] (opacity, r, g, b)
{
    int i = blockIdx.x * blockDim.x + threadIdx.x;
    if (i >= GS_N) return;

    // --- scales / opacity ---
    float s0 = __expf(scales_raw[3*i+0]);
    float s1 = __expf(scales_raw[3*i+1]);
    float s2 = __expf(scales_raw[3*i+2]);
    float opacity = 1.0f / (1.0f + __expf(-opac_raw[i]));

    // --- quaternion -> rotation ---
    float qr = rot_raw[4*i+0], qx = rot_raw[4*i+1], qy = rot_raw[4*i+2], qz = rot_raw[4*i+3];
    float qn = rsqrtf(qr*qr + qx*qx + qy*qy + qz*qz);
    qr *= qn; qx *= qn; qy *= qn; qz *= qn;
    float R[9];
    R[0] = 1.f - 2.f*(qy*qy + qz*qz); R[1] = 2.f*(qx*qy - qr*qz);       R[2] = 2.f*(qx*qz + qr*qy);
    R[3] = 2.f*(qx*qy + qr*qz);       R[4] = 1.f - 2.f*(qx*qx + qz*qz); R[5] = 2.f*(qy*qz - qr*qx);
    R[6] = 2.f*(qx*qz - qr*qy);       R[7] = 2.f*(qy*qz + qr*qx);       R[8] = 1.f - 2.f*(qx*qx + qy*qy);

    // Sigma = (R*diag(s)) (R*diag(s))^T  =>  S[a][b] = sum_c R[a][c]R[b][c] s[c]^2
    float s2v[3] = { s0*s0, s1*s1, s2*s2 };
    float S[3][3];
    #pragma unroll
    for (int a = 0; a < 3; ++a)
        #pragma unroll
        for (int b = 0; b < 3; ++b)
            S[a][b] = R[a*3+0]*R[b*3+0]*s2v[0] + R[a*3+1]*R[b*3+1]*s2v[1] + R[a*3+2]*R[b*3+2]*s2v[2];

    // --- projection ---
    float m0 = means3D[3*i+0], m1 = means3D[3*i+1], m2 = means3D[3*i+2];
    float pv[4], ph[4];
    #pragma unroll
    for (int r = 0; r < 4; ++r) {
        pv[r] = view[r*4+0]*m0 + view[r*4+1]*m1 + view[r*4+2]*m2 + view[r*4+3];
        ph[r] = proj[r*4+0]*m0 + proj[r*4+1]*m1 + proj[r*4+2]*m2 + proj[r*4+3];
    }
    float depth = pv[2];
    float pw = 1.0f / (ph[3] + 1e-7f);
    float ndc0 = ph[0]*pw, ndc1 = ph[1]*pw;
    float mx = ((ndc0 + 1.0f) * (float)GS_W - 1.0f) * 0.5f;
    float my = ((ndc1 + 1.0f) * (float)GS_H - 1.0f) * 0.5f;

    // --- EWA 2D covariance ---
    const float fx = (float)GS_W / (2.0f * GS_TANFOV);
    const float fy = (float)GS_H / (2.0f * GS_TANFOV);
    const float lim = 1.3f * GS_TANFOV;
    float tz = depth;
    float tx = fminf(lim, fmaxf(-lim, pv[0]/tz)) * tz;
    float ty = fminf(lim, fmaxf(-lim, pv[1]/tz)) * tz;
    float J00 = fx/tz, J02 = -fx*tx/(tz*tz);
    float J11 = fy/tz, J12 = -fy*ty/(tz*tz);
    float M0[3], M1[3];
    #pragma unroll
    for (int c = 0; c < 3; ++c) {
        M0[c] = J00*view[0*4+c] + J02*view[2*4+c];
        M1[c] = J11*view[1*4+c] + J12*view[2*4+c];
    }
    float SM0[3], SM1[3];
    #pragma unroll
    for (int a = 0; a < 3; ++a) {
        SM0[a] = S[a][0]*M0[0] + S[a][1]*M0[1] + S[a][2]*M0[2];
        SM1[a] = S[a][0]*M1[0] + S[a][1]*M1[1] + S[a][2]*M1[2];
    }
    float cov00 = M0[0]*SM0[0] + M0[1]*SM0[1] + M0[2]*SM0[2];
    float cov01 = M0[0]*SM1[0] + M0[1]*SM1[1] + M0[2]*SM1[2];
    float cov11 = M1[0]*SM1[0] + M1[1]*SM1[1] + M1[2]*SM1[2];
    float a = cov00 + 0.3f, b = cov01, c = cov11 + 0.3f;
    float det = a*c - b*b;
    float invdet = 1.0f / det;
    float conA = c*invdet, conB = -b*invdet, conC = a*invdet;

    // --- SH color ---
    float dx = m0 - campos[0], dy = m1 - campos[1], dz = m2 - campos[2];
    float dinv = rsqrtf(dx*dx + dy*dy + dz*dz);
    float x = dx*dinv, y = dy*dinv, z = dz*dinv;
    float xx = x*x, yy = y*y, zz = z*z, xy = x*y, yz = y*z, xz = x*z;
    const float C0f = 0.28209479177387814f, C1f = 0.4886025119029199f;
    const float C20 = 1.0925484305920792f,  C21 = -1.0925484305920792f;
    const float C22 = 0.31539156525252005f, C23 = -1.0925484305920792f, C24 = 0.5462742152960396f;
    const float C30 = -0.5900435899266435f, C31 = 2.890611442640554f,  C32 = -0.4570457994644658f;
    const float C33 = 0.3731763325901154f,  C34 = -0.4570457994644658f;
    const float C35 = 1.445305721320277f,   C36 = -0.5900435899266435f;
    const float* sh = shs + (size_t)i * 48;
    float col[3];
    #pragma unroll
    for (int ch = 0; ch < 3; ++ch) {
        float res = C0f * sh[0*3+ch];
        res += -C1f*y*sh[1*3+ch] + C1f*z*sh[2*3+ch] - C1f*x*sh[3*3+ch];
        res += C20*xy*sh[4*3+ch] + C21*yz*sh[5*3+ch] + C22*(2.f*zz-xx-yy)*sh[6*3+ch]
             + C23*xz*sh[7*3+ch] + C24*(xx-yy)*sh[8*3+ch];
        res += C30*y*(3.f*xx-yy)*sh[9*3+ch] + C31*xy*z*sh[10*3+ch]
             + C32*y*(4.f*zz-xx-yy)*sh[11*3+ch] + C33*z*(2.f*zz-3.f*xx-3.f*yy)*sh[12*3+ch]
             + C34*x*(4.f*zz-xx-yy)*sh[13*3+ch] + C35*z*(xx-yy)*sh[14*3+ch]
             + C36*x*(xx-3.f*yy)*sh[15*3+ch];
        col[ch] = fmaxf(res + 0.5f, 0.0f);
    }

    // --- power-form coefficients: power = c0*gx^2 + c1*gy^2 + c2*gx*gy + c3*gx + c4*gy + c5 ---
    bool valid = (depth > 0.2f) && (det > 0.0f);
    float c0, c1, c2, c3, c4, c5;
    if (valid) {
        c0 = -0.5f * conA;
        c1 = -0.5f * conC;
        c2 = -conB;
        c3 = conA*mx + conB*my;
        c4 = conC*my + conB*mx;
        c5 = -0.5f*(conA*mx*mx + conC*my*my) - conB*mx*my;
    } else {
        c0 = c1 = c2 = c3 = c4 = 0.0f;
        c5 = -1e30f;   // exp -> 0, alpha -> 0
    }

    depthKey[i] = depth;
    float* cw = coeff8 + (size_t)i * 8;
    cw[0] = c0; cw[1] = c1; cw[2] = c2; cw[3] = c3; cw[4] = c4; cw[5] = c5;
    cw[6] = 0.0f; cw[7] = 0.0f;
    colOp[i] = make_float4(opacity, col[0], col[1], col[2]);
}

// ---------------------------------------------------------------------------
// Kernel 2: bitonic argsort by depth in LDS + gather into depth order
// ---------------------------------------------------------------------------
__global__ __launch_bounds__(1024) void gs_sort_gather(
    const float* __restrict__ depthKey,
    const float* __restrict__ coeff8,
    const float4* __restrict__ colOp,
    float* __restrict__ sCoeff8,
    float4* __restrict__ sColOp)
{
    __shared__ float skey[GS_N];
    __shared__ int   sidx[GS_N];
    const int tid = threadIdx.x;

    for (int i = tid; i < GS_N; i += 1024) { skey[i] = depthKey[i]; sidx[i] = i; }
    __syncthreads();

    for (int k = 2; k <= GS_N; k <<= 1) {
        for (int j = k >> 1; j > 0; j >>= 1) {
            for (int i = tid; i < GS_N; i += 1024) {
                int ixj = i ^ j;
                if (ixj > i) {
                    bool up = ((i & k) == 0);
                    float ki = skey[i], kj = skey[ixj];
                    bool swap = up ? (ki > kj) : (ki < kj);
                    if (swap) {
                        skey[i] = kj; skey[ixj] = ki;
                        int t = sidx[i]; sidx[i] = sidx[ixj]; sidx[ixj] = t;
                    }
                }
            }
            __syncthreads();
        }
    }

    for (int i = tid; i < GS_N; i += 1024) {
        int s = sidx[i];
        #pragma unroll
        for (int kk = 0; kk < 8; ++kk)
            sCoeff8[(size_t)i*8 + kk] = coeff8[(size_t)s*8 + kk];
        sColOp[i] = colOp[s];
    }
}

// ---------------------------------------------------------------------------
// Kernel 3: rasterize. One wave32 = 16 pixels; per 16-Gaussian chunk the power
// matrix power[16g x 16p] is computed with two chained V_WMMA_F32_16X16X4_F32,
// then blended front-to-back (8 Gaussians per lane half + one xor-16 shuffle).
// ---------------------------------------------------------------------------
__global__ __launch_bounds__(256) void gs_rasterize(
    const float* __restrict__ sCoeff8,  // sorted [N,8]
    const float4* __restrict__ sColOp,  // sorted [N]
    const float* __restrict__ bg,       // [3]
    float* __restrict__ out)            // [3,H,W]
{
    const int lane  = threadIdx.x & 31;
    const int wave  = threadIdx.x >> 5;
    const int strip = blockIdx.x * 8 + wave;           // 576 strips of 16 px
    const int row   = strip / (GS_W / 16);
    const int col0  = (strip % (GS_W / 16)) * 16;
    const int pix   = lane & 15;
    const int half  = lane >> 4;                       // 0: gaussians 0-7, 1: 8-15
    const float gx  = (float)(col0 + pix);
    const float gy  = (float)row;

    // B operand (pixel features), K-major per the 16x4 f32 WMMA layout:
    //  vgpr0: K0 (lanes 0-15) / K2 (lanes 16-31); vgpr1: K1 / K3
    v2f b1, b2;
    if (half == 0) { b1.x = gx*gx;  b1.y = gy*gy; b2.x = gy;  b2.y = 1.0f; }
    else           { b1.x = gx*gy;  b1.y = gx;    b2.x = 0.f; b2.y = 0.f;  }

    float Tacc = 1.0f, Cr = 0.0f, Cg = 0.0f, Cb = 0.0f;

    for (int chunk = 0; chunk < GS_N/16; ++chunk) {
        const int gbase = chunk * 16;
        const float* cp = sCoeff8 + (size_t)(gbase + pix) * 8;

        // A operand (gaussian coefficients): lane = gaussian row, halves carry K pairs
        v2f a1, a2;
        a1.x = cp[half*2 + 0];
        a1.y = cp[half*2 + 1];
        if (half == 0) { a2.x = cp[4]; a2.y = cp[5]; }
        else           { a2.x = 0.f;   a2.y = 0.f;   }

        if (chunk + 1 < GS_N/16)
            __builtin_prefetch(sCoeff8 + (size_t)(gbase + 16 + pix) * 8, 0, 1);

        v8f pw = {0.f,0.f,0.f,0.f,0.f,0.f,0.f,0.f};
        pw = __builtin_amdgcn_wmma_f32_16x16x4_f32(false, a1, false, b1, (short)0, pw, false, false);
        pw = __builtin_amdgcn_wmma_f32_16x16x4_f32(false, a2, false, b2, (short)0, pw, false, false);

        // local front-to-back blend over this half's 8 gaussians
        const float4* co = sColOp + gbase + half*8;
        float T = 1.0f, lr = 0.0f, lg = 0.0f, lb = 0.0f;
        #pragma unroll
        for (int t = 0; t < 8; ++t) {
            float4 c4 = co[t];
            float p = pw[t];
            float alpha = fminf(0.99f, c4.x * __expf(p));
            if (p > 0.0f || alpha < (1.0f/255.0f)) alpha = 0.0f;
            float w = T * alpha;
            lr += w * c4.y; lg += w * c4.z; lb += w * c4.w;
            T *= (1.0f - alpha);
        }

        // stitch halves: low half (gaussians 0-7) composites before high half
        float orr = __shfl_xor(lr, 16, 32);
        float org = __shfl_xor(lg, 16, 32);
        float orb = __shfl_xor(lb, 16, 32);
        float orT = __shfl_xor(T,  16, 32);
        float lowR  = (half == 0) ? lr : orr;
        float lowG  = (half == 0) ? lg : org;
        float lowB  = (half == 0) ? lb : orb;
        float lowT  = (half == 0) ? T  : orT;
        float highR = (half == 0) ? orr : lr;
        float highG = (half == 0) ? org : lg;
        float highB = (half == 0) ? orb : lb;
        float highT = (half == 0) ? orT : T;

        float cRr = lowR + lowT * highR;
        float cRg = lowG + lowT * highG;
        float cRb = lowB + lowT * highB;
        float cT  = lowT * highT;

        Cr += Tacc * cRr;
        Cg += Tacc * cRg;
        Cb += Tacc * cRb;
        Tacc *= cT;

        if (__all(Tacc < 1e-7f)) break;   // wave-uniform early termination
    }

    if (half == 0) {
        int p = row * GS_W + col0 + pix;
        out[0*GS_H*GS_W + p] = Cr + Tacc * bg[0];
        out[1*GS_H*GS_W + p] = Cg + Tacc * bg[1];
        out[2*GS_H*GS_W + p] = Cb + Tacc * bg[2];
    }
}

// ---------------------------------------------------------------------------
extern "C" void kernel_launch(void* const* d_in, const int* in_sizes, int n_in,
                              void* d_out, int out_size, void* d_ws, size_t ws_size,
                              hipStream_t stream)
{
    const float* means  = (const float*)d_in[0];
    const float* scales = (const float*)d_in[1];
    const float* rots   = (const float*)d_in[2];
    const float* opac   = (const float*)d_in[3];
    const float* shs    = (const float*)d_in[4];
    const float* view   = (const float*)d_in[5];
    const float* proj   = (const float*)d_in[6];
    const float* campos = (const float*)d_in[7];
    const float* bg     = (const float*)d_in[8];

    char* ws = (char*)d_ws;
    float*  depthKey = (float*)ws;   ws += GS_N * sizeof(float);
    float*  coeff8   = (float*)ws;   ws += GS_N * 8 * sizeof(float);
    float4* colOp    = (float4*)ws;  ws += GS_N * sizeof(float4);
    float*  sCoeff8  = (float*)ws;   ws += GS_N * 8 * sizeof(float);
    float4* sColOp   = (float4*)ws;  ws += GS_N * sizeof(float4);

    gs_preprocess<<<GS_N/256, 256, 0, stream>>>(means, scales, rots, opac, shs,
                                                view, proj, campos,
                                                depthKey, coeff8, colOp);
    gs_sort_gather<<<1, 1024, 0, stream>>>(depthKey, coeff8, colOp, sCoeff8, sColOp);
    gs_rasterize<<<(GS_H*GS_W/16)/8, 256, 0, stream>>>(sCoeff8, sColOp, bg, (float*)d_out);
}